// ExtractionModel_69226282877186
// MI455X (gfx1250) — compile-verified
//
#include <hip/hip_runtime.h>
#include <stdint.h>
#include <math.h>

#define Hh 192
#define Ww 192
#define Cc 256
#define HWp (Hh*Ww)
#define NCAND (3*HWp)
#define MAXF 2048
#define SORT_N 4096

typedef float v2f __attribute__((ext_vector_type(2)));
typedef float v8f __attribute__((ext_vector_type(8)));

// zero-padded score read (matches F.conv2d padding=1)
__device__ __forceinline__ float sAt(const float* __restrict__ s, int i, int j) {
    return (i >= 0 && i < Hh && j >= 0 && j < Ww) ? s[i * Ww + j] : 0.0f;
}

struct Loc { float det; float ki, kj; int valid; };

// NMS + handcrafted localization (Newton step) for one pixel
__device__ __forceinline__ Loc localize(const float* __restrict__ sm, int i, int j) {
    float c = sm[i * Ww + j];
    // 3x3 max, -inf padding (reduce_window semantics)
    float m = c;
    #pragma unroll
    for (int di = -1; di <= 1; ++di)
        #pragma unroll
        for (int dj = -1; dj <= 1; ++dj) {
            int ii = i + di, jj = j + dj;
            if (ii >= 0 && ii < Hh && jj >= 0 && jj < Ww) m = fmaxf(m, sm[ii * Ww + jj]);
        }
    float det = (m == c) ? c : 0.0f;

    float a00 = sAt(sm, i-1, j-1), a01 = sAt(sm, i-1, j), a02 = sAt(sm, i-1, j+1);
    float a10 = sAt(sm, i,   j-1),                        a12 = sAt(sm, i,   j+1);
    float a20 = sAt(sm, i+1, j-1), a21 = sAt(sm, i+1, j), a22 = sAt(sm, i+1, j+1);

    float di_ = 0.5f * (a21 - a01);
    float dj_ = 0.5f * (a12 - a10);
    float dii = a01 - 2.0f * c + a21;
    float djj = a10 - 2.0f * c + a12;
    float dij = 0.25f * (a00 - a02 - a20 + a22);
    float dd = dii * djj - dij * dij;
    if (dd == 0.0f) dd = 1e-20f;
    float stepi = -(djj * di_ - dij * dj_) / dd;
    float stepj = -(-dij * di_ + dii * dj_) / dd;

    int valid = (det != 0.0f) && (fabsf(stepi) < 0.5f) && (fabsf(stepj) < 0.5f);
    float ki = valid ? ((float)i + stepi) : 0.0f;
    float kj = valid ? ((float)j + stepj) : 0.0f;
    float i0 = floorf(ki), j0 = floorf(kj);
    valid = valid && (i0 >= 0.0f) && (j0 >= 0.0f) &&
            (i0 + 1.0f <= (float)(Hh - 1)) && (j0 + 1.0f <= (float)(Ww - 1));
    Loc r; r.det = det; r.ki = ki; r.kj = kj; r.valid = valid;
    return r;
}

// ---- Stage 1: det maps + padded candidate scores -------------------------
__global__ void k_stage1(const float* __restrict__ s0, const float* __restrict__ s1,
                         const float* __restrict__ s2, float* __restrict__ detOut,
                         float* __restrict__ cand) {
    int idx = blockIdx.x * blockDim.x + threadIdx.x;
    if (idx >= NCAND) return;
    int scale = idx / HWp, rem = idx - scale * HWp;
    int i = rem / Ww, j = rem - i * Ww;
    const float* sm = (scale == 0) ? s0 : (scale == 1) ? s1 : s2;
    Loc L = localize(sm, i, j);
    detOut[idx] = L.det;
    cand[idx] = L.valid ? L.det : -INFINITY;
}

// ---- Stage 2: streaming bitonic top-2048 (deterministic, matches top_k) --
__global__ void k_topk(const float* __restrict__ cand,
                       unsigned long long* __restrict__ selKeys) {
    __shared__ unsigned long long keys[SORT_N];
    const int tid = threadIdx.x;
    for (int t = tid; t < MAXF; t += blockDim.x) keys[t] = ~0ull;
    __syncthreads();
    const int nchunks = NCAND / MAXF;  // 54
    for (int ch = 0; ch < nchunks; ++ch) {
        for (int t = tid; t < MAXF; t += blockDim.x) {
            int idx = ch * MAXF + t;
            unsigned u = __float_as_uint(cand[idx]);
            unsigned ord = (u & 0x80000000u) ? ~u : (u | 0x80000000u);  // ascending-orderable
            keys[MAXF + t] = ((unsigned long long)(~ord) << 32) | (unsigned)idx;
        }
        __syncthreads();
        // full bitonic sort ascending: low 2048 = best (desc score, asc idx)
        for (unsigned k = 2; k <= SORT_N; k <<= 1) {
            for (unsigned j = k >> 1; j > 0; j >>= 1) {
                for (unsigned ii = tid; ii < SORT_N; ii += blockDim.x) {
                    unsigned ixj = ii ^ j;
                    if (ixj > ii) {
                        unsigned long long a = keys[ii], b = keys[ixj];
                        bool up = ((ii & k) == 0);
                        if ((a > b) == up) { keys[ii] = b; keys[ixj] = a; }
                    }
                }
                __syncthreads();
            }
        }
    }
    for (int t = tid; t < MAXF; t += blockDim.x) selKeys[t] = keys[t];
}

// ---- Stage 3: per-keypoint descriptors via block-diagonal WMMA GEMM ------
// One wave (32 lanes) per block handles 16 keypoints x 256 channels.
__global__ void k_extract(const float* __restrict__ f0, const float* __restrict__ f1,
                          const float* __restrict__ f2, const float* __restrict__ cand,
                          const unsigned long long* __restrict__ selKeys,
                          float* __restrict__ outKp, float* __restrict__ outDesc,
                          float* __restrict__ outSc,
                          const float* __restrict__ s0, const float* __restrict__ s1,
                          const float* __restrict__ s2) {
    __shared__ float sDesc[16 * Cc];
    __shared__ float sW00[16], sW01[16], sW10[16], sW11[16];
    __shared__ int   sI0[16], sJ0[16], sScale[16];
    __shared__ float sInv[16];

    const int l = threadIdx.x;         // 0..31
    const int base = blockIdx.x * 16;  // output slot base

    if (l < 16) {
        int t = base + l;
        unsigned long long key = selKeys[t];
        unsigned idx = (unsigned)(key & 0xffffffffu);
        float sc = cand[idx];
        bool sel = (sc != -INFINITY);
        int scale = idx / HWp, rem = idx - scale * HWp;
        int i = rem / Ww, j = rem - i * Ww;
        const float* sm = (scale == 0) ? s0 : (scale == 1) ? s1 : s2;
        Loc L = localize(sm, i, j);
        int i0 = min(max((int)floorf(L.ki), 0), Hh - 2);
        int j0 = min(max((int)floorf(L.kj), 0), Ww - 2);
        float wi = L.ki - (float)i0, wj = L.kj - (float)j0;
        float w00 = (1.0f - wi) * (1.0f - wj), w01 = (1.0f - wi) * wj;
        float w10 = wi * (1.0f - wj),          w11 = wi * wj;
        if (!sel) { w00 = w01 = w10 = w11 = 0.0f; i0 = 0; j0 = 0; }
        sW00[l] = w00; sW01[l] = w01; sW10[l] = w10; sW11[l] = w11;
        sI0[l] = i0; sJ0[l] = j0; sScale[l] = scale;
        outKp[2 * t]     = sel ? (L.kj * 16.0f + 7.5f) : 0.0f;  // x
        outKp[2 * t + 1] = sel ? (L.ki * 16.0f + 7.5f) : 0.0f;  // y
        outSc[t] = sel ? sc : 0.0f;
    }
    __syncthreads();

    const bool hi = (l >= 16);
    const int  ln = l & 15;
    float sq = 0.0f;

    for (int ct = 0; ct < 16; ++ct) {     // 16 channel tiles of 16
        v8f acc = {0, 0, 0, 0, 0, 0, 0, 0};
        #pragma unroll
        for (int p = 0; p < 16; ++p) {    // K=4 slice per keypoint (block-diagonal B)
            int scale = sScale[p];
            const float* fb = (scale == 0) ? f0 : (scale == 1) ? f1 : f2;
            int c = ct * 16 + ln;                       // A row = channel
            int row = sI0[p] + (hi ? 1 : 0);            // taps (i0,*) or (i0+1,*)
            const float* addr = fb + (size_t)c * HWp + row * Ww + sJ0[p];
            v2f a; a.x = addr[0]; a.y = addr[1];        // K pair: (w00,w01) row or (w10,w11) row
            float bw0 = hi ? sW10[p] : sW00[p];
            float bw1 = hi ? sW11[p] : sW01[p];
            v2f b; b.x = (ln == p) ? bw0 : 0.0f;        // B column p only
            v2f bb = b; bb.y = (ln == p) ? bw1 : 0.0f; bb.x = b.x;
            acc = __builtin_amdgcn_wmma_f32_16x16x4_f32(
                false, a, false, bb, (short)0, acc, false, false);
        }
        #pragma unroll
        for (int r = 0; r < 8; ++r) {     // D: lane<16 -> M=r, lane>=16 -> M=8+r
            float v = acc[r];
            sq += v * v;
            sDesc[ln * Cc + ct * 16 + (hi ? 8 + r : r)] = v;
        }
    }

    float tot = sq + __shfl_xor(sq, 16, 32);
    if (!hi) sInv[ln] = 1.0f / fmaxf(sqrtf(tot), 1e-12f);
    __syncthreads();

    for (int e = l; e < 16 * Cc; e += 32) {
        int kp = e >> 8, c = e & 255;
        outDesc[(size_t)(base + kp) * Cc + c] = sDesc[e] * sInv[kp];
    }
}

extern "C" void kernel_launch(void* const* d_in, const int* in_sizes, int n_in,
                              void* d_out, int out_size, void* d_ws, size_t ws_size,
                              hipStream_t stream) {
    const float* fe = (const float*)d_in[0];
    const float* fm = (const float*)d_in[1];
    const float* fd = (const float*)d_in[2];
    const float* se = (const float*)d_in[3];
    const float* sm = (const float*)d_in[4];
    const float* sd = (const float*)d_in[5];

    float* out     = (float*)d_out;
    float* outKp   = out;                          // [2048,2]
    float* outDesc = out + MAXF * 2;               // [2048,256]
    float* outSc   = outDesc + (size_t)MAXF * Cc;  // [2048]
    float* outDet  = outSc + MAXF;                 // 3 x [192,192]

    float* cand = (float*)d_ws;                                            // 3*H*W floats
    unsigned long long* selKeys =
        (unsigned long long*)((char*)d_ws + (size_t)NCAND * sizeof(float)); // 2048 u64

    k_stage1<<<(NCAND + 255) / 256, 256, 0, stream>>>(se, sm, sd, outDet, cand);
    k_topk<<<1, 1024, 0, stream>>>(cand, selKeys);
    k_extract<<<MAXF / 16, 32, 0, stream>>>(fe, fm, fd, cand, selKeys,
                                            outKp, outDesc, outSc, se, sm, sd);
}